// SNNLIF_63187558858840
// MI455X (gfx1250) — compile-verified
//
#include <hip/hip_runtime.h>

typedef float v2f __attribute__((ext_vector_type(2)));
typedef float v8f __attribute__((ext_vector_type(8)));

#define N_IN    12
#define N_HID   32
#define N_OUT   3
#define BATCH   8192
#define TSTEPS  256
#define ALPHA   0.9375f
#define THRESH  1.0f

#define WAVES_PER_BLOCK 2
#define THREADS_PER_BLOCK (WAVES_PER_BLOCK * 32)
#define LDS_STRIDE 18   // 16 batch cols + pad; 16*18 % 64 == 32 -> conflict-free reads

__global__ __launch_bounds__(THREADS_PER_BLOCK)
void snn_lif_wmma(const float* __restrict__ x,    // [B, T, N_IN]
                  const float* __restrict__ W1,   // [N_HID, N_IN]
                  const float* __restrict__ W2,   // [N_OUT, N_HID]
                  float* __restrict__ out)        // [B, N_OUT]
{
    __shared__ float sh_lds[WAVES_PER_BLOCK][N_HID][LDS_STRIDE];

    const int tid  = threadIdx.x;
    const int lane = tid & 31;
    const int wIn  = tid >> 5;                          // wave index in block
    const int wave = blockIdx.x * WAVES_PER_BLOCK + wIn;
    const int bb   = wave * 16;                         // 16 batch rows per wave
    const int lmod = lane & 15;
    const int hh   = lane >> 4;                         // lane half (K-half / hidden-half)
    const int b    = bb + lmod;                         // this lane's batch element

    // ---- W1 in WMMA B-layout: B(k,n) -> lane = n + 16*(k>=2), vgpr = k&1.
    //      w1b[c][t].x = W1[n][4c+2hh], .y = W1[n][4c+2hh+1], n = t*16 + lmod
    v2f w1b[3][2];
#pragma unroll
    for (int c = 0; c < 3; ++c) {
#pragma unroll
        for (int t2 = 0; t2 < 2; ++t2) {
            const int n = t2 * 16 + lmod;
            const int k = 4 * c + 2 * hh;
            v2f w; w.x = W1[n * N_IN + k]; w.y = W1[n * N_IN + k + 1];
            w1b[c][t2] = w;
        }
    }

    // ---- W2 per-lane: this lane covers hidden half hh of batch b
    float w2r[16][N_OUT];
#pragma unroll
    for (int j = 0; j < 16; ++j)
#pragma unroll
        for (int o = 0; o < N_OUT; ++o)
            w2r[j][o] = W2[o * N_HID + hh * 16 + j];

    v8f vh0 = {0.f,0.f,0.f,0.f,0.f,0.f,0.f,0.f};   // hidden potential, N-tile 0 (C/D layout)
    v8f vh1 = {0.f,0.f,0.f,0.f,0.f,0.f,0.f,0.f};   // hidden potential, N-tile 1
    float vo[N_OUT]  = {0.f, 0.f, 0.f};
    float cnt[N_OUT] = {0.f, 0.f, 0.f};

    const long xrow = (long)b * TSTEPS;

    for (int t = 0; t < TSTEPS; ++t) {
        // ---- A operands in WMMA A-layout: A(m,k) -> lane = m + 16*(k>=2), vgpr = k&1
        const float* xp = x + (xrow + t) * N_IN + 2 * hh;
        __builtin_prefetch(xp + 8 * N_IN, 0, 0);        // stream ahead (global_prefetch_b8)
        const v2f a0 = *reinterpret_cast<const v2f*>(xp + 0);
        const v2f a1 = *reinterpret_cast<const v2f*>(xp + 4);
        const v2f a2 = *reinterpret_cast<const v2f*>(xp + 8);

        // ---- layer 1: vh = alpha*vh + x_t @ W1^T  (alpha-decay folded into C)
        v8f acc0 = vh0 * ALPHA;
        v8f acc1 = vh1 * ALPHA;
        acc0 = __builtin_amdgcn_wmma_f32_16x16x4_f32(false, a0, false, w1b[0][0], (short)0, acc0, false, false);
        acc0 = __builtin_amdgcn_wmma_f32_16x16x4_f32(false, a1, false, w1b[1][0], (short)0, acc0, false, false);
        acc0 = __builtin_amdgcn_wmma_f32_16x16x4_f32(false, a2, false, w1b[2][0], (short)0, acc0, false, false);
        acc1 = __builtin_amdgcn_wmma_f32_16x16x4_f32(false, a0, false, w1b[0][1], (short)0, acc1, false, false);
        acc1 = __builtin_amdgcn_wmma_f32_16x16x4_f32(false, a1, false, w1b[1][1], (short)0, acc1, false, false);
        acc1 = __builtin_amdgcn_wmma_f32_16x16x4_f32(false, a2, false, w1b[2][1], (short)0, acc1, false, false);

        // ---- spike + reset; element i of tile tN is (batch m = i + 8*hh, hidden n = tN*16 + lmod)
        float shv0[8], shv1[8];
#pragma unroll
        for (int i = 0; i < 8; ++i) {
            const bool f0 = acc0[i] >= THRESH;
            const bool f1 = acc1[i] >= THRESH;
            shv0[i] = f0 ? 1.f : 0.f;  acc0[i] = f0 ? 0.f : acc0[i];
            shv1[i] = f1 ? 1.f : 0.f;  acc1[i] = f1 ? 0.f : acc1[i];
        }
        vh0 = acc0; vh1 = acc1;

        // ---- stage sh into LDS as [hidden][batch] (wave-private slice; DS ops are
        //      in-order per wave, so no barrier needed — just pin compile-time order)
        float* p0 = &sh_lds[wIn][lmod     ][8 * hh];
        float* p1 = &sh_lds[wIn][16 + lmod][8 * hh];
#pragma unroll
        for (int i = 0; i < 8; i += 2) {
            v2f s0; s0.x = shv0[i]; s0.y = shv0[i + 1];
            v2f s1; s1.x = shv1[i]; s1.y = shv1[i + 1];
            *reinterpret_cast<v2f*>(p0 + i) = s0;
            *reinterpret_cast<v2f*>(p1 + i) = s1;
        }
        asm volatile("" ::: "memory");

        // ---- layer 2: per-lane dot over 16 hidden (half hh) for batch b
        float po[N_OUT] = {0.f, 0.f, 0.f};
#pragma unroll
        for (int j = 0; j < 16; ++j) {
            const float sv = sh_lds[wIn][hh * 16 + j][lmod];
#pragma unroll
            for (int o = 0; o < N_OUT; ++o)
                po[o] = fmaf(sv, w2r[j][o], po[o]);
        }
        asm volatile("" ::: "memory");

        // combine the two hidden halves across lane halves
#pragma unroll
        for (int o = 0; o < N_OUT; ++o)
            po[o] += __shfl_xor(po[o], 16, 32);

        // ---- output LIF: vo = alpha*vo + sh@W2^T; spike, reset, count
#pragma unroll
        for (int o = 0; o < N_OUT; ++o) {
            const float v = ALPHA * vo[o] + po[o];
            const bool  f = v >= THRESH;
            cnt[o] += f ? 1.f : 0.f;
            vo[o]   = f ? 0.f : v;
        }
    }

    if (lane < 16) {
#pragma unroll
        for (int o = 0; o < N_OUT; ++o)
            out[(long)b * N_OUT + o] = cnt[o];
    }
}

extern "C" void kernel_launch(void* const* d_in, const int* in_sizes, int n_in,
                              void* d_out, int out_size, void* d_ws, size_t ws_size,
                              hipStream_t stream) {
    const float* x  = (const float*)d_in[0];   // [8192, 256, 12]
    const float* W1 = (const float*)d_in[1];   // [32, 12]
    const float* W2 = (const float*)d_in[2];   // [3, 32]
    float* out = (float*)d_out;                // [8192, 3]

    const int waves  = BATCH / 16;                       // 512 waves
    const int blocks = waves / WAVES_PER_BLOCK;          // 256 blocks
    snn_lif_wmma<<<blocks, THREADS_PER_BLOCK, 0, stream>>>(x, W1, W2, out);
}